// HausdorffERLoss_87136296501857
// MI455X (gfx1250) — compile-verified
//
#include <hip/hip_runtime.h>

// ---------------------------------------------------------------------------
// HausdorffER loss, MI455X (gfx1250). Bandwidth-bound stencil:
//   ~3.1 GB total traffic -> ~135us floor at 23.3 TB/s.
// CDNA5 paths used: global_load_async_to_lds_b128 (ASYNCcnt) tile staging,
// s_wait_asynccnt, wave32 reductions.
// ---------------------------------------------------------------------------

#define WIMG 1024
#define HIMG 1024
#define BATCH 32
#define WP 1032                 // padded width: 4 left + 1024 + 4 right
#define HP 1026                 // padded height: 1 top + 1024 + 1 bottom
#define SLICE ((size_t)WP * (size_t)HP)
#define TX 128
#define TY 32
#define LDS_W 136               // 34 b128 chunks per row
#define LDS_ROWS 34
#define NCHUNK (LDS_ROWS * 34)  // 1156 b128 transfers per tile
#define ITERS 10
#define NELEM_F ((float)(BATCH) * (float)(HIMG) * (float)(WIMG))  // 33554432
#define HW_F ((float)(HIMG) * (float)(WIMG))                      // 1048576

__device__ __forceinline__ void async_tile_load_b128(unsigned lds_addr,
                                                     const float* gptr) {
  // copies 16 bytes from the global address in gptr to the LDS address in lds_addr, tracked by ASYNCcnt.
  asm volatile("global_load_async_to_lds_b128 %0, %1, off"
               :
               : "v"(lds_addr), "v"(gptr)
               : "memory");
}

__device__ __forceinline__ void wait_asynccnt0() {
  asm volatile("s_wait_asynccnt 0x0" ::: "memory");
}

// ---------------------------------------------------------------------------
// Zero the pad borders of both ping-pong arrays (they stay zero: stencil
// kernels only write interior). Zero border == conv zero-padding.
// ---------------------------------------------------------------------------
#define PER_B (2 * WP + HIMG * 8)       // 10256 pad elements / batch / array
#define PER_ARR (BATCH * PER_B)         // 328192

__global__ __launch_bounds__(256) void hd_zero_pads(float* Ea, float* Eb) {
  int idx = blockIdx.x * 256 + threadIdx.x;
  if (idx >= 2 * PER_ARR) return;
  float* E = (idx < PER_ARR) ? Ea : Eb;
  int i = idx % PER_ARR;
  int b = i / PER_B;
  int j = i % PER_B;
  size_t off;
  if (j < 2 * WP) {                       // top & bottom pad rows
    int row = (j < WP) ? 0 : (HP - 1);
    int col = j % WP;
    off = (size_t)row * WP + col;
  } else {                                // left/right 4-wide pad columns
    int t = j - 2 * WP;
    int row = 1 + t / 8;
    int cc = t % 8;
    int col = (cc < 4) ? cc : (1024 + cc);  // cols 0..3 and 1028..1031
    off = (size_t)row * WP + col;
  }
  E[(size_t)b * SLICE + off] = 0.0f;
}

// ---------------------------------------------------------------------------
// Init per-iteration, per-batch stats {sum, max, min}: 10 x 32 x 3 floats.
// max/min are stored as uint-punned nonnegative floats for lock-free atomics.
// ---------------------------------------------------------------------------
__global__ void hd_init_stats(float* stats) {
  int i = blockIdx.x * blockDim.x + threadIdx.x;
  if (i < ITERS * BATCH) {
    stats[i * 3 + 0] = 0.0f;
    ((unsigned*)stats)[i * 3 + 1] = 0u;           // max init: 0.0f
    ((unsigned*)stats)[i * 3 + 2] = 0x7F800000u;  // min init: +inf
  }
}

// ---------------------------------------------------------------------------
// bound = (pred - (target==0))^2, written into padded E_a interior.
// ---------------------------------------------------------------------------
__global__ __launch_bounds__(256) void hd_init_bound(
    const float* __restrict__ pred, const int* __restrict__ tgt,
    float* __restrict__ Ea) {
  size_t idx = (size_t)blockIdx.x * 256 + threadIdx.x;  // quad index
  int xq = (int)(idx & 255);
  size_t rest = idx >> 8;
  int y = (int)(rest & 1023);
  int b = (int)(rest >> 10);
  size_t gin = (((size_t)b * HIMG) + y) * WIMG + (size_t)xq * 4;
  float4 p = *(const float4*)(pred + gin);
  int4 t = *(const int4*)(tgt + gin);
  float4 o;
  o.x = p.x - ((t.x == 0) ? 1.0f : 0.0f); o.x *= o.x;
  o.y = p.y - ((t.y == 0) ? 1.0f : 0.0f); o.y *= o.y;
  o.z = p.z - ((t.z == 0) ? 1.0f : 0.0f); o.z *= o.z;
  o.w = p.w - ((t.w == 0) ? 1.0f : 0.0f); o.w *= o.w;
  *(float4*)(Ea + (size_t)b * SLICE + (size_t)(y + 1) * WP + (size_t)xq * 4 + 4) = o;
}

// ---------------------------------------------------------------------------
// One erosion iteration, fully fused:
//   e_out = max( a*S - bc*N - 0.5, 0 )
// where S = 5-tap sum of unnormalized e_in (zero border), N = in-bounds tap
// count, and (a, bc) analytically apply the previous iteration's per-batch
// min/max normalization: a = 0.2/denom, bc = 0.2*emin/denom.
// Also reduces per-batch {sum, max, min} of e_out for this iteration.
// ---------------------------------------------------------------------------
__global__ __launch_bounds__(256) void hd_erosion_step(
    const float* __restrict__ Ein, float* __restrict__ Eout,
    const float* __restrict__ prevStats, float* __restrict__ curStats,
    int first) {
  __shared__ float tile[LDS_ROWS * LDS_W];
  __shared__ float red[3 * 8];

  const int tid = threadIdx.x;
  const int x0 = blockIdx.x * TX;
  const int y0 = blockIdx.y * TY;
  const int b = blockIdx.z;

  const float* src = Ein + (size_t)b * SLICE;
  unsigned lds_base = (unsigned)(size_t)(&tile[0]);

  // Stage 34x136 halo tile into LDS via async b128 copies (no VGPR bounce).
  for (int c = tid; c < NCHUNK; c += 256) {
    int r = c / 34;
    int q = c - r * 34;
    const float* g = src + (size_t)(y0 + r) * WP + (size_t)(x0 + 4 * q);
    async_tile_load_b128(lds_base + (unsigned)((r * LDS_W + 4 * q) * 4), g);
  }

  // Previous iteration's normalization constants (identity on first iter).
  float a, bc;
  if (first) {
    a = 0.2f;
    bc = 0.0f;
  } else {
    float mx = prevStats[b * 3 + 1];
    float mn = prevStats[b * 3 + 2];
    float denom = mx - mn;
    float inv = (denom != 0.0f) ? 1.0f / denom : 1.0f;
    float emin = (denom != 0.0f) ? mn : 0.0f;
    a = 0.2f * inv;
    bc = 0.2f * emin * inv;
  }

  wait_asynccnt0();
  __syncthreads();

  const int lane = tid & 31;
  const int ty0 = (tid >> 5) * 4;
  float* dst = Eout + (size_t)b * SLICE;

  float lsum = 0.0f, lmax = 0.0f, lmin = __builtin_inff();

#pragma unroll
  for (int r = 0; r < 4; ++r) {
    int ly = ty0 + r;
    int y = y0 + ly;
    int row = (ly + 1) * LDS_W;
    float nY = 5.0f - (float)(y == 0) - (float)(y == HIMG - 1);
#pragma unroll
    for (int i = 0; i < 4; ++i) {
      int lx = lane + 32 * i;  // lane-interleaved: conflict-free LDS, coalesced stores
      int x = x0 + lx;
      int col = lx + 4;
      float S = tile[row + col] + tile[row + col - 1] + tile[row + col + 1] +
                tile[row - LDS_W + col] + tile[row + LDS_W + col];
      float N = nY - (float)(x == 0) - (float)(x == WIMG - 1);
      float e = fmaxf(fmaf(a, S, -fmaf(bc, N, 0.5f)), 0.0f);
      dst[(size_t)(y + 1) * WP + (size_t)(x + 4)] = e;
      lsum += e;
      lmax = fmaxf(lmax, e);
      lmin = fminf(lmin, e);
    }
  }

  // wave32 reduction
#pragma unroll
  for (int off = 16; off > 0; off >>= 1) {
    lsum += __shfl_xor(lsum, off, 32);
    lmax = fmaxf(lmax, __shfl_xor(lmax, off, 32));
    lmin = fminf(lmin, __shfl_xor(lmin, off, 32));
  }
  int wv = tid >> 5;
  if (lane == 0) {
    red[wv] = lsum;
    red[8 + wv] = lmax;
    red[16 + wv] = lmin;
  }
  __syncthreads();
  if (tid == 0) {
    float s = red[0], mx = red[8], mn = red[16];
#pragma unroll
    for (int w = 1; w < 8; ++w) {
      s += red[w];
      mx = fmaxf(mx, red[8 + w]);
      mn = fminf(mn, red[16 + w]);
    }
    atomicAdd(&curStats[b * 3 + 0], s);
    // values are >= 0, so uint ordering == float ordering
    atomicMax((unsigned int*)&curStats[b * 3 + 1], __float_as_uint(mx));
    atomicMin((unsigned int*)&curStats[b * 3 + 2], __float_as_uint(mn));
  }
}

// ---------------------------------------------------------------------------
// loss = (1/(B*H*W)) * sum_b sum_k (k+1)^2 * [(sum_k - HW*min_k)/denom_k]
// (or unnormalized sum if denom == 0). One wave32; lane b handles batch b.
// ---------------------------------------------------------------------------
__global__ void hd_finalize(const float* __restrict__ stats,
                            float* __restrict__ out) {
  int b = threadIdx.x;
  float acc = 0.0f;
  if (b < BATCH) {
    for (int k = 0; k < ITERS; ++k) {
      const float* s = stats + (size_t)(k * BATCH + b) * 3;
      float sum = s[0], mx = s[1], mn = s[2];
      float denom = mx - mn;
      float w = (float)((k + 1) * (k + 1));
      float Snorm = (denom != 0.0f) ? (sum - HW_F * mn) / denom : sum;
      acc += w * Snorm;
    }
  }
#pragma unroll
  for (int off = 16; off > 0; off >>= 1) acc += __shfl_xor(acc, off, 32);
  if (b == 0) out[0] = acc / NELEM_F;
}

// ---------------------------------------------------------------------------
extern "C" void kernel_launch(void* const* d_in, const int* in_sizes, int n_in,
                              void* d_out, int out_size, void* d_ws,
                              size_t ws_size, hipStream_t stream) {
  (void)in_sizes; (void)n_in; (void)out_size; (void)ws_size;
  const float* pred = (const float*)d_in[0];
  const int* target = (const int*)d_in[1];
  float* out = (float*)d_out;

  char* ws = (char*)d_ws;
  float* stats = (float*)ws;                         // 960 floats
  float* Ea = (float*)(ws + 4096);                   // padded state A
  float* Eb = Ea + (size_t)BATCH * SLICE;            // padded state B

  hd_zero_pads<<<(2 * PER_ARR + 255) / 256, 256, 0, stream>>>(Ea, Eb);
  hd_init_stats<<<(ITERS * BATCH + 255) / 256, 256, 0, stream>>>(stats);

  // bound -> Ea
  hd_init_bound<<<(BATCH * HIMG * (WIMG / 4)) / 256, 256, 0, stream>>>(
      pred, target, Ea);

  dim3 grid(WIMG / TX, HIMG / TY, BATCH);  // (8, 32, 32)
  for (int k = 0; k < ITERS; ++k) {
    const float* ein = (k & 1) ? Eb : Ea;
    float* eout = (k & 1) ? Ea : Eb;
    const float* prev = (k == 0) ? stats : (stats + (size_t)(k - 1) * BATCH * 3);
    float* cur = stats + (size_t)k * BATCH * 3;
    hd_erosion_step<<<grid, 256, 0, stream>>>(ein, eout, prev, cur, k == 0);
  }

  hd_finalize<<<1, 32, 0, stream>>>(stats, out);
}